// S2Convolution_21947282882656
// MI455X (gfx1250) — compile-verified
//
#include <hip/hip_runtime.h>

// ---------------------------------------------------------------------------
// Problem constants
// ---------------------------------------------------------------------------
#define A_ 2
#define B_ 2
#define N_ 1024
#define C_ 8
#define ITEMS 32768            // A*B*N*C
#define NTRI 42

typedef __attribute__((ext_vector_type(2))) float v2f;
typedef __attribute__((ext_vector_type(8))) float v8f;

// ---------------------------------------------------------------------------
// Triple tables (l, l1, l2) in reference order — constexpr for template meta
// ---------------------------------------------------------------------------
constexpr int c_l [NTRI] = {0,0,0,0,0,
                            2,2,2,2,2,2,2,2,
                            4,4,4,4,4,4,4,4,4,4,
                            6,6,6,6,6,6,6,6,6,6,
                            8,8,8,8,8,8,8,8,8};
constexpr int c_l1[NTRI] = {0,2,4,6,8,
                            0,2,2,4,4,6,6,8,
                            0,2,2,2,4,4,4,6,6,8,
                            0,2,2,2,4,4,4,6,6,8,
                            0,2,2,4,4,4,6,6,8};
constexpr int c_l2[NTRI] = {0,2,4,6,8,
                            2,2,4,4,6,6,8,8,
                            4,2,4,6,4,6,8,6,8,8,
                            6,4,6,8,4,6,8,6,8,8,
                            8,6,8,4,6,8,6,8,8};

constexpr int al4(int x) { return (x + 3) & ~3; }
constexpr int dsz(int l) { return 2 * l + 1; }
constexpr int cg_size(int t) { return dsz(c_l[t]) * dsz(c_l1[t]) * dsz(c_l2[t]); }
constexpr int cg_off(int t) { int o = 0; for (int i = 0; i < t; ++i) o += cg_size(i); return o; }
constexpr int CG_HALF = cg_off(NTRI);          // 52334
constexpr int CG_TOTAL = 2 * CG_HALF;
static_assert(CG_HALF == 52334, "triple table mismatch");

// Padded CG layouts: R[t] = (Dl*D1) x al4(D2), L[t] = (Dl*D2) x al4(D1)
constexpr int padR_size(int t) { return dsz(c_l[t]) * dsz(c_l1[t]) * al4(dsz(c_l2[t])); }
constexpr int padL_size(int t) { return dsz(c_l[t]) * dsz(c_l2[t]) * al4(dsz(c_l1[t])); }
constexpr int padR_off(int t) { int o = 0; for (int i = 0; i < t; ++i) o += padR_size(i); return o; }
constexpr int PADR_TOTAL = padR_off(NTRI);
constexpr int padL_off(int t) { int o = PADR_TOTAL; for (int i = 0; i < t; ++i) o += padL_size(i); return o; }
constexpr int PAD_TOTAL = padL_off(NTRI);

constexpr size_t rh_off_c(int l) {
  return l == 0 ? 0 : l == 2 ? 32768 : l == 4 ? 851968 : l == 6 ? 3506176 : (size_t)9043968;
}
constexpr size_t rhn_out_off(int l) {
  return l == 0 ? 163840 : l == 2 ? 196608 : l == 4 ? 1015808 : l == 6 ? 3670016 : 9207808;
}
constexpr int tstart(int l) { return l == 0 ? 0 : l == 2 ? 5 : l == 4 ? 13 : l == 6 ? 23 : 33; }
constexpr int tend(int l)   { return l == 0 ? 5 : l == 2 ? 13 : l == 4 ? 23 : l == 6 ? 33 : 42; }
constexpr int KZMAX = al4(289);  // padded row stride of the z-operand LDS tiles

// Runtime copies for the (non-templated) pad kernel
__device__ const int d_tl0[NTRI] = {0,0,0,0,0, 2,2,2,2,2,2,2,2, 4,4,4,4,4,4,4,4,4,4,
                                    6,6,6,6,6,6,6,6,6,6, 8,8,8,8,8,8,8,8,8};
__device__ const int d_tl1[NTRI] = {0,2,4,6,8, 0,2,2,4,4,6,6,8, 0,2,2,2,4,4,4,6,6,8,
                                    0,2,2,2,4,4,4,6,6,8, 0,2,2,4,4,4,6,6,8};
__device__ const int d_tl2[NTRI] = {0,2,4,6,8, 2,2,4,4,6,6,8,8, 4,2,4,6,4,6,8,6,8,8,
                                    6,4,6,8,4,6,8,6,8,8, 8,6,8,4,6,8,6,8,8};

__device__ float g_CG[CG_TOTAL];     // compact CG tables (numpy order)
__device__ float g_CGpad[PAD_TOTAL]; // K-padded, GEMM-ready layouts

// ---------------------------------------------------------------------------
// CDNA5 async global->LDS copy (ASYNCcnt-tracked), per CDNA5 ISA + 07.
// ---------------------------------------------------------------------------
__device__ __forceinline__ void async_ld_f32(float* lds_dst, const float* g) {
  auto l3 = (__attribute__((address_space(3))) float*)lds_dst;  // LDS offset in one VGPR
  asm volatile("global_load_async_to_lds_b32 %0, %1, off"
               :: "v"(l3), "v"(g)
               : "memory");
}
__device__ __forceinline__ void wait_async0() {
  asm volatile("s_wait_asynccnt 0x0" ::: "memory");
}

// ---------------------------------------------------------------------------
// Kernel 0: regenerate CG tables = numpy RandomState(42).randn(...)*0.1 (f32)
// MT19937 + legacy polar gauss with caching, 53-bit doubles — bit-exact.
// ---------------------------------------------------------------------------
__global__ void init_cg_kernel() {
  if (threadIdx.x != 0 || blockIdx.x != 0) return;
  unsigned mt[624];
  int mti;
  mt[0] = 42u;
  for (int i = 1; i < 624; ++i)
    mt[i] = 1812433253u * (mt[i - 1] ^ (mt[i - 1] >> 30)) + (unsigned)i;
  mti = 624;
  double gcache = 0.0;
  int has_g = 0;

  auto gen = [&]() -> unsigned {
    if (mti >= 624) {
      for (int k = 0; k < 624; ++k) {
        unsigned y = (mt[k] & 0x80000000u) | (mt[(k + 1) % 624] & 0x7fffffffu);
        mt[k] = mt[(k + 397) % 624] ^ (y >> 1) ^ ((y & 1u) ? 0x9908b0dfu : 0u);
      }
      mti = 0;
    }
    unsigned y = mt[mti++];
    y ^= y >> 11;
    y ^= (y << 7) & 0x9d2c5680u;
    y ^= (y << 15) & 0xefc60000u;
    y ^= y >> 18;
    return y;
  };
  auto rdbl = [&]() -> double {
    unsigned a = gen() >> 5, b = gen() >> 6;
    return (a * 67108864.0 + b) / 9007199254740992.0;
  };
  auto gauss = [&]() -> double {
    if (has_g) { has_g = 0; return gcache; }
    double x1, x2, r2;
    do {
      x1 = 2.0 * rdbl() - 1.0;
      x2 = 2.0 * rdbl() - 1.0;
      r2 = x1 * x1 + x2 * x2;
    } while (r2 >= 1.0 || r2 == 0.0);
    double f = sqrt(-2.0 * log(r2) / r2);
    gcache = f * x1;
    has_g = 1;
    return f * x2;
  };
  for (int i = 0; i < CG_TOTAL; ++i) g_CG[i] = (float)(gauss() * 0.1);
}

// ---------------------------------------------------------------------------
// Kernel 0b: rewrite CG into K-padded GEMM layouts (zero-filled pad columns)
// blockIdx: [0,42) -> R side, [42,84) -> L side
// ---------------------------------------------------------------------------
__global__ void pad_cg_kernel() {
  const int side = blockIdx.x / NTRI;  // 0 = R, 1 = L
  const int t = blockIdx.x % NTRI;
  int srcOff = 0, dstOff = (side == 0) ? 0 : PADR_TOTAL;
  for (int i = 0; i < t; ++i) {
    const int Dl = 2 * d_tl0[i] + 1, D1 = 2 * d_tl1[i] + 1, D2 = 2 * d_tl2[i] + 1;
    srcOff += Dl * D1 * D2;
    dstOff += (side == 0) ? Dl * D1 * ((D2 + 3) & ~3) : Dl * D2 * ((D1 + 3) & ~3);
  }
  const int Dl = 2 * d_tl0[t] + 1, D1 = 2 * d_tl1[t] + 1, D2 = 2 * d_tl2[t] + 1;
  const int M = (side == 0) ? Dl * D1 : Dl * D2;
  const int K = (side == 0) ? D2 : D1;
  const int Kp = (K + 3) & ~3;
  const float* src = g_CG + ((side == 0) ? 0 : CG_HALF) + srcOff;
  float* dst = g_CGpad + dstOff;
  for (int idx = threadIdx.x; idx < M * Kp; idx += blockDim.x) {
    const int row = idx / Kp, col = idx % Kp;
    dst[idx] = (col < K) ? src[row * K + col] : 0.f;
  }
}

// ---------------------------------------------------------------------------
// Kernel 1: spectral conv  rh[l][item][r][k] = sum_i x[..,i,r]*w[..,i,c,k] (+bias)
// ---------------------------------------------------------------------------
template <int L>
__global__ void spectral_kernel(const float* __restrict__ x,
                                const float* __restrict__ w,
                                const float* __restrict__ bias,
                                float* __restrict__ rh) {
  constexpr int DL = 2 * L + 1;
  constexpr int IL = (L == 0) ? 5 : 2;
  const long total = (long)ITEMS * DL * DL;
  long idx = (long)blockIdx.x * blockDim.x + threadIdx.x;
  if (idx >= total) return;
  int item = (int)(idx / (DL * DL));
  int rk = (int)(idx % (DL * DL));
  int r = rk / DL, k = rk % DL;
  int c = item & (C_ - 1);
  int n = (item >> 3) & (N_ - 1);
  int ab = item >> 13;
  float s = 0.f;
#pragma unroll
  for (int i = 0; i < IL; ++i)
    s += x[((long)(ab * N_ + n) * IL + i) * DL + r] *
         w[((long)(ab * IL + i) * C_ + c) * DL + k];
  if (L == 0) s += bias[ab * C_ + c];
  rh[(size_t)item * DL * DL + rk] = s;
}

// ---------------------------------------------------------------------------
// Kernel 2: fused CG tensor product. One block = one item. Fully
// compile-time-specialized per triple (template recursion), all operand loads
// unguarded (K zero-padded, M/N index-clamped), b64-friendly layouts,
// rh tiles staged via CDNA5 async global->LDS.
// ---------------------------------------------------------------------------
__device__ __forceinline__ int imin(int a, int b) { return a < b ? a : b; }

template <int LOUT, int T>
__device__ __forceinline__ void process_triple(int tid, int wave, int lane,
                                               const float* __restrict__ rh, int item,
                                               float* s_X, float* s_Y,
                                               float* s_bX, float* s_bY,
                                               float* s_raw1, float* s_raw2,
                                               v8f& zacc, int zmt, int znt, int zkc) {
  constexpr int DLV = 2 * LOUT + 1;
  constexpr int D1 = 2 * c_l1[T] + 1;
  constexpr int D2 = 2 * c_l2[T] + 1;
  constexpr int KKZ = D1 * D2;
  constexpr int KKZP = al4(KKZ);
  constexpr int KXp = al4(D2);          // K-pad of the XX GEMM
  constexpr int KYp = al4(D1);          // K-pad of the YY GEMM
  constexpr int MT = (DLV + 15) / 16;
  constexpr int ZT = MT * MT;
  constexpr int KW = 8 / ZT;

  const int lan16 = lane & 15;
  const int khalf = (lane < 16) ? 0 : 2;

  __syncthreads();  // previous triple's readers of s_* are done

  // ---- Async-stage raw rh blocks into LDS (wave 0), others zero s_X/s_Y K-pad
  {
    const float* rh1g = rh + rh_off_c(c_l1[T]) + (size_t)item * (D1 * D1);
    const float* rh2g = rh + rh_off_c(c_l2[T]) + (size_t)item * (D2 * D2);
    if (wave == 0) {
      for (int i = lane; i < D1 * D1; i += 32) async_ld_f32(&s_raw1[i], &rh1g[i]);
      for (int i = lane; i < D2 * D2; i += 32) async_ld_f32(&s_raw2[i], &rh2g[i]);
      wait_async0();
    }
    if constexpr (KKZP > KKZ) {
      for (int i = tid; i < DLV * (KKZP - KKZ); i += 256) {
        const int r = i / (KKZP - KKZ), cc = KKZ + i % (KKZP - KKZ);
        s_X[r * KZMAX + cc] = 0.f;
        s_Y[r * KZMAX + cc] = 0.f;
      }
    }
  }
  __syncthreads();

  // ---- Pack K-padded B tiles from the raw LDS blocks
  for (int i = tid; i < D2 * KXp; i += 256) {       // s_bX[q][j2] = rh2[q,j2]
    const int q = i / KXp, j2 = i % KXp;
    s_bX[i] = (j2 < D2) ? s_raw2[q * D2 + j2] : 0.f;
  }
  for (int i = tid; i < KYp * D1; i += 256) {       // s_bY[j1][p] = rh1[j1,p]
    const int j1 = i / D1;
    s_bY[i] = (j1 < D1) ? s_raw1[i] : 0.f;
  }
  __syncthreads();

  // ---- Phase 1: XX[(j,p),q] and YY[(k,q),p] GEMMs (shared CG as A-matrix)
  constexpr int MX = DLV * D1, NX = D2;
  constexpr int MY = DLV * D2, NY = D1;
  constexpr int mtX = (MX + 15) / 16, ntX = (NX + 15) / 16;
  constexpr int mtY = (MY + 15) / 16, ntY = (NY + 15) / 16;
  constexpr int UX = mtX * ntX, UY = mtY * ntY;
  const float* cgR = g_CGpad + padR_off(T);
  const float* cgL = g_CGpad + padL_off(T);

  for (int u = wave; u < UX; u += 8) {  // XX: A=CG_R (MX x KXp), B=s_bX^T
    const int mtile = u / ntX, ntile = u % ntX;
    const int row0 = imin(mtile * 16 + lan16, MX - 1);
    const int col0 = imin(ntile * 16 + lan16, NX - 1);
    v8f acc = {0.f, 0.f, 0.f, 0.f, 0.f, 0.f, 0.f, 0.f};
#pragma unroll
    for (int k0 = 0; k0 < KXp; k0 += 4) {
      v2f a = *(const v2f*)&cgR[row0 * KXp + k0 + khalf];
      v2f b = *(const v2f*)&s_bX[col0 * KXp + k0 + khalf];
      acc = __builtin_amdgcn_wmma_f32_16x16x4_f32(false, a, false, b, (short)0,
                                                  acc, false, false);
    }
#pragma unroll
    for (int v = 0; v < 8; ++v) {  // scatter into z-ready layout s_X[j][p*D2+q]
      const int rr = mtile * 16 + v + ((lane >= 16) ? 8 : 0);
      const int cc = ntile * 16 + lan16;
      if (rr < MX && cc < NX) {
        const int j = rr / D1, p = rr % D1;
        s_X[j * KZMAX + p * D2 + cc] = acc[v];
      }
    }
  }
  for (int u = wave; u < UY; u += 8) {  // YY: A=CG_L (MY x KYp), B=s_bY
    const int mtile = u / ntY, ntile = u % ntY;
    const int row0 = imin(mtile * 16 + lan16, MY - 1);
    const int col0 = imin(ntile * 16 + lan16, NY - 1);
    v8f acc = {0.f, 0.f, 0.f, 0.f, 0.f, 0.f, 0.f, 0.f};
#pragma unroll
    for (int k0 = 0; k0 < KYp; k0 += 4) {
      v2f a = *(const v2f*)&cgL[row0 * KYp + k0 + khalf];
      v2f b;
      b.x = s_bY[(k0 + khalf) * D1 + col0];
      b.y = s_bY[(k0 + khalf + 1) * D1 + col0];
      acc = __builtin_amdgcn_wmma_f32_16x16x4_f32(false, a, false, b, (short)0,
                                                  acc, false, false);
    }
#pragma unroll
    for (int v = 0; v < 8; ++v) {  // scatter into z-ready layout s_Y[k][p*D2+q]
      const int rr = mtile * 16 + v + ((lane >= 16) ? 8 : 0);
      const int cc = ntile * 16 + lan16;
      if (rr < MY && cc < NY) {
        const int k = rr / D2, q = rr % D2;
        s_Y[k * KZMAX + cc * D2 + q] = acc[v];
      }
    }
  }
  __syncthreads();

  // ---- Phase 2: z[k,j] += sum_m Y'[k,m] X'[j,m]; K split across KW waves
  {
    constexpr int ksteps = KKZP / 4;
    constexpr int chunk = (ksteps + KW - 1) / KW;
    const int kbeg = zkc * chunk * 4;
    int kend = kbeg + chunk * 4;
    if (kend > KKZP) kend = KKZP;
    const int arow = imin(zmt * 16 + lan16, DLV - 1);
    const int bcol = imin(znt * 16 + lan16, DLV - 1);
    const float* Abase = &s_Y[arow * KZMAX + khalf];
    const float* Bbase = &s_X[bcol * KZMAX + khalf];
#pragma unroll 4
    for (int m0 = kbeg; m0 < kend; m0 += 4) {
      v2f a = *(const v2f*)&Abase[m0];
      v2f b = *(const v2f*)&Bbase[m0];
      zacc = __builtin_amdgcn_wmma_f32_16x16x4_f32(false, a, false, b, (short)0,
                                                   zacc, false, false);
    }
  }
}

template <int LOUT, int T, int TEND>
__device__ __forceinline__ void triple_seq(int tid, int wave, int lane,
                                           const float* __restrict__ rh, int item,
                                           float* s_X, float* s_Y,
                                           float* s_bX, float* s_bY,
                                           float* s_raw1, float* s_raw2,
                                           v8f& zacc, int zmt, int znt, int zkc) {
  process_triple<LOUT, T>(tid, wave, lane, rh, item, s_X, s_Y, s_bX, s_bY,
                          s_raw1, s_raw2, zacc, zmt, znt, zkc);
  if constexpr (T + 1 < TEND)
    triple_seq<LOUT, T + 1, TEND>(tid, wave, lane, rh, item, s_X, s_Y, s_bX, s_bY,
                                  s_raw1, s_raw2, zacc, zmt, znt, zkc);
}

template <int LOUT>
__global__ __launch_bounds__(256) void couple_kernel(const float* __restrict__ rh,
                                                     float* __restrict__ out) {
  constexpr int DL = 2 * LOUT + 1;
  constexpr int MT = (DL + 15) / 16;
  constexpr int ZT = MT * MT;

  __shared__ float s_X[DL * KZMAX];   // X'[j][m], K-padded
  __shared__ float s_Y[DL * KZMAX];   // Y'[k][m], K-padded
  __shared__ float s_bX[17 * 20];
  __shared__ float s_bY[20 * 17];
  __shared__ float s_raw1[289];       // async-staged rh1
  __shared__ float s_raw2[289];       // async-staged rh2
  __shared__ float s_z[DL * DL];
  __shared__ float s_feat;

  const int tid = threadIdx.x;
  const int wave = tid >> 5;
  const int lane = tid & 31;
  const int item = blockIdx.x;
  const int c = item & (C_ - 1);
  const int n = (item >> 3) & (N_ - 1);
  const int ab = item >> 13;

  for (int i = tid; i < DL * DL; i += 256) s_z[i] = 0.f;
  if (tid == 0) s_feat = 0.f;

  v8f zacc = {0.f, 0.f, 0.f, 0.f, 0.f, 0.f, 0.f, 0.f};
  const int ztile = wave % ZT;
  const int zmt = ztile / MT;
  const int znt = ztile % MT;
  const int zkc = wave / ZT;

  triple_seq<LOUT, tstart(LOUT), tend(LOUT)>(tid, wave, lane, rh, item,
                                             s_X, s_Y, s_bX, s_bY, s_raw1, s_raw2,
                                             zacc, zmt, znt, zkc);

  __syncthreads();
#pragma unroll
  for (int v = 0; v < 8; ++v) {
    const int rr = zmt * 16 + v + ((lane >= 16) ? 8 : 0);
    const int cc = znt * 16 + (lane & 15);
    if (rr < DL && cc < DL) atomicAdd(&s_z[rr * DL + cc], zacc[v]);
  }
  __syncthreads();

  float* rhn = out + rhn_out_off(LOUT) + (size_t)item * DL * DL;
  float part = 0.f;
  for (int i = tid; i < DL * DL; i += 256) {
    const float zv = s_z[i];
    rhn[i] = zv;
    part += zv * zv;
  }
  atomicAdd(&s_feat, part);
  __syncthreads();
  if (tid == 0) {
    const float nl = (float)(78.95683520871486 / (double)(2 * LOUT + 1));  // 8*pi^2/(2l+1)
    out[(size_t)ab * 40960 + (LOUT / 2) * 8192 + n * 8 + c] = nl * s_feat;
  }
}

// ---------------------------------------------------------------------------
// Launch
// ---------------------------------------------------------------------------
extern "C" void kernel_launch(void* const* d_in, const int* in_sizes, int n_in,
                              void* d_out, int out_size, void* d_ws, size_t ws_size,
                              hipStream_t stream) {
  const float* x0 = (const float*)d_in[0];
  const float* w0 = (const float*)d_in[1];
  const float* x2 = (const float*)d_in[2];
  const float* w2 = (const float*)d_in[3];
  const float* x4 = (const float*)d_in[4];
  const float* w4 = (const float*)d_in[5];
  const float* x6 = (const float*)d_in[6];
  const float* w6 = (const float*)d_in[7];
  const float* x8 = (const float*)d_in[8];
  const float* w8 = (const float*)d_in[9];
  const float* bias = (const float*)d_in[10];
  float* out = (float*)d_out;
  float* rh = (float*)d_ws;  // needs 18,513,920 floats (~74 MB)

  init_cg_kernel<<<1, 1, 0, stream>>>();
  pad_cg_kernel<<<2 * NTRI, 256, 0, stream>>>();

  spectral_kernel<0><<<(ITEMS * 1 + 255) / 256, 256, 0, stream>>>(x0, w0, bias, rh + 0);
  spectral_kernel<2><<<(ITEMS * 25 + 255) / 256, 256, 0, stream>>>(x2, w2, bias, rh + 32768);
  spectral_kernel<4><<<(ITEMS * 81 + 255) / 256, 256, 0, stream>>>(x4, w4, bias, rh + 851968);
  spectral_kernel<6><<<(ITEMS * 169 + 255) / 256, 256, 0, stream>>>(x6, w6, bias, rh + 3506176);
  spectral_kernel<8><<<(ITEMS * 289 + 255) / 256, 256, 0, stream>>>(x8, w8, bias, rh + 9043968);

  couple_kernel<0><<<ITEMS, 256, 0, stream>>>(rh, out);
  couple_kernel<2><<<ITEMS, 256, 0, stream>>>(rh, out);
  couple_kernel<4><<<ITEMS, 256, 0, stream>>>(rh, out);
  couple_kernel<6><<<ITEMS, 256, 0, stream>>>(rh, out);
  couple_kernel<8><<<ITEMS, 256, 0, stream>>>(rh, out);
}